// GAT_76020921140371
// MI455X (gfx1250) — compile-verified
//
#include <hip/hip_runtime.h>
#include <hip/hip_bf16.h>

typedef __attribute__((ext_vector_type(2))) float v2f;
typedef __attribute__((ext_vector_type(8))) float v8f;
typedef __attribute__((ext_vector_type(4))) unsigned u32x4;
typedef __attribute__((ext_vector_type(8))) int i32x8;
typedef __attribute__((ext_vector_type(4))) int i32x4;

#define NEG_SLOPE 0.2f
#define EPS_DEN 1e-16f

// ---------- helpers ----------

// Order-preserving float -> uint map so segment-max can use native u32 atomic max.
// Identity of the mapped max is 0u, so plain zero-memset initializes the buffers.
__device__ __forceinline__ unsigned f2ord(float f) {
  unsigned u = __float_as_uint(f);
  return (u & 0x80000000u) ? ~u : (u | 0x80000000u);
}
__device__ __forceinline__ float ord2f(unsigned u) {
  return __uint_as_float((u & 0x80000000u) ? (u ^ 0x80000000u) : ~u);
}

__device__ __forceinline__ void atomicAddF(float* p, float v) {
  (void)__hip_atomic_fetch_add(p, v, __ATOMIC_RELAXED, __HIP_MEMORY_SCOPE_AGENT);
}

__device__ __forceinline__ float leaky(float x) {
  return x > 0.f ? x : NEG_SLOPE * x;
}

// ---------- Tensor Data Mover: DMA a [tile_h x tile_w] f32 tile into LDS ----------
// D# per CDNA5 ISA ch.8: group0 = {count, lds_addr, global_addr, type=2},
// group1 = {data_size=4B, tensor dims/strides, tile dims}, groups 2/3 zero (2D).
// This toolchain exposes the 6-arg builtin (extra trailing int32x8 group).
__device__ __forceinline__ void tdm_load_tile_2d_f32(
    unsigned lds_addr, const float* gptr, unsigned tensor_w, unsigned tensor_h,
    unsigned tile_w, unsigned tile_h, unsigned row_stride_elems) {
  unsigned long long ga = (unsigned long long)(size_t)gptr;
  u32x4 g0;
  g0[0] = 1u;                                    // count=1, user descriptor
  g0[1] = lds_addr;                              // LDS byte address of tile start
  g0[2] = (unsigned)(ga & 0xffffffffu);          // global_addr[31:0]
  g0[3] = (unsigned)((ga >> 32) & 0x01ffffffu)   // global_addr[56:32]
          | (2u << 30);                          // type=2 ("image")
  i32x8 g1;
  g1[0] = (int)(2u << 16);                       // data_size=2 -> 4 bytes; mask/flags 0
  g1[1] = (int)((tensor_w & 0xffffu) << 16);     // tensor_dim0[15:0] in [31:16]
  g1[2] = (int)(((tensor_w >> 16) & 0xffffu) |   // tensor_dim0[31:16]
                ((tensor_h & 0xffffu) << 16));   // tensor_dim1[15:0]
  g1[3] = (int)(((tensor_h >> 16) & 0xffffu) |   // tensor_dim1[31:16]
                ((tile_w & 0xffffu) << 16));     // tile_dim0
  g1[4] = (int)(tile_h & 0xffffu);               // tile_dim1 (tile_dim2=0)
  g1[5] = (int)row_stride_elems;                 // tensor_dim0_stride[31:0]
  g1[6] = 0;                                     // stride[47:32]=0, dim1_stride lo=0
  g1[7] = 0;                                     // dim1_stride hi=0
  i32x4 gz4 = {0, 0, 0, 0};
  i32x8 gz8 = {0, 0, 0, 0, 0, 0, 0, 0};
  __builtin_amdgcn_tensor_load_to_lds(g0, g1, gz4, gz4, gz8, 0);
}

// ---------- fp32 WMMA GEMM with TDM-staged B: C[M,Nc] = A[M,K] * B[K,Nc] ----------
// Block = 5 waves; blockIdx.y selects a 16-column slab of B that the TDM stages
// into LDS once per block; each wave computes one 16x16 output tile.
#define GWAVES 5
__global__ __launch_bounds__(32 * GWAVES) void gemm_f32_wmma_tdm(
    const float* __restrict__ A, const float* __restrict__ B,
    float* __restrict__ C, int M, int K, int Nc) {
  __shared__ float ldsB[256 * 16];               // up to K=256 rows x 16 cols (16KB)
  int wid = threadIdx.x >> 5;
  int lane = threadIdx.x & 31;
  int c0 = blockIdx.y << 4;

  if (wid == 0) {
    tdm_load_tile_2d_f32((unsigned)(size_t)ldsB, B + c0,
                         (unsigned)Nc, (unsigned)K, 16u, (unsigned)K,
                         (unsigned)Nc);
    __builtin_amdgcn_s_wait_tensorcnt(0);
  }
  __syncthreads();                               // LDS tile visible to all 5 waves

  int tr = blockIdx.x * GWAVES + wid;
  int r0 = tr << 4;
  if (r0 < M) {                                  // wave-uniform (EXEC stays full)
    int lm = lane & 15;
    int kh = (lane >> 4) << 1;                   // lanes 0-15: K{k,k+1}; 16-31: {k+2,k+3}
    int rA = r0 + lm; if (rA >= M) rA = M - 1;   // clamp (M=50000 is exact anyway)
    const float* arow = A + (size_t)rA * K + kh;
    const float* brow = ldsB + kh * 16 + lm;

    v8f acc = {};
    for (int k = 0; k < K; k += 4) {
      v2f a, b;
      a[0] = arow[0];
      a[1] = arow[1];
      b[0] = brow[0];
      b[1] = brow[16];
      arow += 4;
      brow += 4 * 16;
      acc = __builtin_amdgcn_wmma_f32_16x16x4_f32(
          false, a, false, b, (short)0, acc, false, false);
    }
    // C/D layout: VGPR i -> M = r0 + i (+8 for lanes 16-31), N = c0 + (lane&15)
    int rbase = r0 + ((lane >> 4) << 3);
    float* crow = C + (size_t)rbase * Nc + c0 + lm;
#pragma unroll
    for (int i = 0; i < 8; i++) {
      if (rbase + i < M) crow[(size_t)i * Nc] = acc[i];
    }
  }
}

// ---------- per-node attention scalars: as[n,h]=<h_row, a_src[h]>, ad likewise ----------
template <int H, int D>
__global__ __launch_bounds__(256) void node_alpha(
    const float* __restrict__ h, const float* __restrict__ a_src,
    const float* __restrict__ a_dst, float* __restrict__ as,
    float* __restrict__ ad, int N) {
  int t = blockIdx.x * blockDim.x + threadIdx.x;
  if (t >= N * H) return;
  int n = t / H, hd = t - n * H;
  const float* row = h + (size_t)n * (H * D) + hd * D;
  float ss = 0.f, dd = 0.f;
#pragma unroll
  for (int j = 0; j < D; j++) {
    float v = row[j];
    ss += v * a_src[hd * D + j];
    dd += v * a_dst[hd * D + j];
  }
  as[t] = ss;
  ad[t] = dd;
}

// ---------- edge pass 1: segment max of leaky logits (ordered-uint atomic max) ----------
template <int H>
__global__ __launch_bounds__(256) void edge_max(
    const int* __restrict__ src, const int* __restrict__ dst,
    const float* __restrict__ as, const float* __restrict__ ad,
    unsigned* __restrict__ m, int E) {
  int e = blockIdx.x * blockDim.x + threadIdx.x;
  if (e >= E) return;
  int s = src[e], d = dst[e];
#pragma unroll
  for (int hh = 0; hh < H; hh++) {
    float lg = leaky(as[s * H + hh] + ad[d * H + hh]);
    atomicMax(&m[d * H + hh], f2ord(lg));
  }
}

// ---------- edge pass 2: segment sum of exp(e - m[dst]) ----------
template <int H>
__global__ __launch_bounds__(256) void edge_sum(
    const int* __restrict__ src, const int* __restrict__ dst,
    const float* __restrict__ as, const float* __restrict__ ad,
    const unsigned* __restrict__ m, float* __restrict__ sden, int E) {
  int e = blockIdx.x * blockDim.x + threadIdx.x;
  if (e >= E) return;
  int s = src[e], d = dst[e];
#pragma unroll
  for (int hh = 0; hh < H; hh++) {
    float lg = leaky(as[s * H + hh] + ad[d * H + hh]);
    float ex = expf(lg - ord2f(m[d * H + hh]));
    atomicAddF(&sden[d * H + hh], ex);
  }
}

// ---------- edge pass 3: out[dst] += alpha * h[src]; one wave32 per edge ----------
template <int H, int D>
__global__ __launch_bounds__(256) void edge_aggregate(
    const int* __restrict__ src, const int* __restrict__ dst,
    const float* __restrict__ as, const float* __restrict__ ad,
    const unsigned* __restrict__ m, const float* __restrict__ sden,
    const float* __restrict__ hmat, float* __restrict__ out, int E) {
  constexpr int F = H * D;
  constexpr int VEC = F / 32;                    // 4 (layer1), 2 (layer2); D % VEC == 0
  int gwave = (int)((blockIdx.x * blockDim.x + threadIdx.x) >> 5);
  if (gwave >= E) return;
  int lane = threadIdx.x & 31;
  int s = src[gwave], d = dst[gwave];
  int hd = (lane * VEC) / D;                     // head owned by this lane's slice
  float lg = leaky(as[s * H + hd] + ad[d * H + hd]);
  float ex = expf(lg - ord2f(m[d * H + hd]));
  float alpha = ex / (sden[d * H + hd] + EPS_DEN);
  const float* hrow = hmat + (size_t)s * F + lane * VEC;  // coalesced row gather
  float* orow = out + (size_t)d * F + lane * VEC;
#pragma unroll
  for (int j = 0; j < VEC; j++) atomicAddF(&orow[j], alpha * hrow[j]);
}

// ---------- elementwise: h = elu(h + b) in place ----------
__global__ __launch_bounds__(256) void elu_bias_inplace(
    float* __restrict__ h, const float* __restrict__ b, int total, int F) {
  int t = blockIdx.x * blockDim.x + threadIdx.x;
  if (t >= total) return;
  float v = h[t] + b[t & (F - 1)];
  h[t] = v > 0.f ? v : (expf(v) - 1.f);
}

// ---------- elementwise: out += b ----------
__global__ __launch_bounds__(256) void bias_add(
    float* __restrict__ out, const float* __restrict__ b, int total, int F) {
  int t = blockIdx.x * blockDim.x + threadIdx.x;
  if (t >= total) return;
  out[t] += b[t & (F - 1)];
}

// ---------- launch ----------
extern "C" void kernel_launch(void* const* d_in, const int* in_sizes, int n_in,
                              void* d_out, int out_size, void* d_ws, size_t ws_size,
                              hipStream_t stream) {
  constexpr int IN = 256, H1 = 8, D1 = 16, F1 = 128, F2 = 64;
  const float* x   = (const float*)d_in[0];
  const int*   ei  = (const int*)d_in[1];
  const float* W1  = (const float*)d_in[2];
  const float* a1s = (const float*)d_in[3];
  const float* a1d = (const float*)d_in[4];
  const float* b1  = (const float*)d_in[5];
  const float* W2  = (const float*)d_in[6];
  const float* a2s = (const float*)d_in[7];
  const float* a2d = (const float*)d_in[8];
  const float* b2  = (const float*)d_in[9];
  float* out = (float*)d_out;

  const int N = in_sizes[0] / IN;      // 50000
  const int E = in_sizes[1] / 2;       // 1.6M
  const int* src = ei;
  const int* dst = ei + E;

  // workspace partition (all 4-byte types)
  float* h1   = (float*)d_ws;                        // N*F1  (reused as layer-2 GEMM out)
  float* agg1 = h1 + (size_t)N * F1;                 // N*F1  (becomes h2 after ELU)
  float* as1  = agg1 + (size_t)N * F1;               // N*H1
  float* ad1  = as1 + (size_t)N * H1;                // N*H1
  unsigned* m1 = (unsigned*)(ad1 + (size_t)N * H1);  // N*H1
  float* s1   = (float*)(m1 + (size_t)N * H1);       // N*H1
  float* as2  = s1 + (size_t)N * H1;                 // N
  float* ad2  = as2 + N;                             // N
  unsigned* m2 = (unsigned*)(ad2 + N);               // N
  float* s2   = (float*)(m2 + N);                    // N

  const int TB = 256;
  auto blocks = [](long long work, int tb) { return (unsigned)((work + tb - 1) / tb); };

  // zero-init accumulators (ordered-uint max identity is 0 as well)
  (void)hipMemsetAsync(agg1, 0, (size_t)N * F1 * 4, stream);
  (void)hipMemsetAsync(m1, 0, (size_t)N * H1 * 4, stream);
  (void)hipMemsetAsync(s1, 0, (size_t)N * H1 * 4, stream);
  (void)hipMemsetAsync(m2, 0, (size_t)N * 4, stream);
  (void)hipMemsetAsync(s2, 0, (size_t)N * 4, stream);
  (void)hipMemsetAsync(out, 0, (size_t)N * F2 * 4, stream);

  const int mtiles = (N + 15) / 16;
  const unsigned gx = (unsigned)((mtiles + GWAVES - 1) / GWAVES);

  // ===== layer 1 =====
  {
    dim3 g(gx, F1 / 16);
    gemm_f32_wmma_tdm<<<g, 32 * GWAVES, 0, stream>>>(x, W1, h1, N, IN, F1);
  }
  node_alpha<H1, D1><<<blocks((long long)N * H1, TB), TB, 0, stream>>>(h1, a1s, a1d, as1, ad1, N);
  edge_max<H1><<<blocks(E, TB), TB, 0, stream>>>(src, dst, as1, ad1, m1, E);
  edge_sum<H1><<<blocks(E, TB), TB, 0, stream>>>(src, dst, as1, ad1, m1, s1, E);
  edge_aggregate<H1, D1><<<blocks((long long)E * 32, TB), TB, 0, stream>>>(
      src, dst, as1, ad1, m1, s1, h1, agg1, E);
  elu_bias_inplace<<<blocks((long long)N * F1, TB), TB, 0, stream>>>(agg1, b1, N * F1, F1);

  // ===== layer 2 =====  (h2 = agg1; layer-2 features written into h1 buffer)
  float* h3 = h1;
  {
    dim3 g(gx, F2 / 16);
    gemm_f32_wmma_tdm<<<g, 32 * GWAVES, 0, stream>>>(agg1, W2, h3, N, F1, F2);
  }
  node_alpha<1, F2><<<blocks(N, TB), TB, 0, stream>>>(h3, a2s, a2d, as2, ad2, N);
  edge_max<1><<<blocks(E, TB), TB, 0, stream>>>(src, dst, as2, ad2, m2, E);
  edge_sum<1><<<blocks(E, TB), TB, 0, stream>>>(src, dst, as2, ad2, m2, s2, E);
  edge_aggregate<1, F2><<<blocks((long long)E * 32, TB), TB, 0, stream>>>(
      src, dst, as2, ad2, m2, s2, h3, out, E);
  bias_add<<<blocks((long long)N * F2, TB), TB, 0, stream>>>(out, b2, N * F2, F2);
}